// HadamardRotation_79826262163491
// MI455X (gfx1250) — compile-verified
//
#include <hip/hip_runtime.h>

typedef __attribute__((ext_vector_type(2))) float v2f;
typedef __attribute__((ext_vector_type(8))) float v8f;

#define DIM 4096
#define TPB 256
// padded tensor: 16 x 16 x 16 with row stride 17 dwords (bank-conflict-free)
#define TSTRIDE 17
#define TSIZE (16 * 16 * TSTRIDE)

__device__ __forceinline__ float hval(int m, int k) {
    // Sylvester Hadamard: H[m][k] = (-1)^popcount(m & k)
    return (__builtin_popcount(m & k) & 1) ? -1.0f : 1.0f;
}

// One 16x16x16 product T = H16 * S done as 4 chained V_WMMA_F32_16X16X4_F32.
// S[k][n] and D[m][n] live at t[base + idx*sK + n*sN] (D overwrites S in place;
// LDS ops from the same wave are in-order, and tiles are wave-private per stage).
__device__ __forceinline__ void had_tile(float* t, int base, int sK, int sN,
                                         int half, int n16, const v2f* aH) {
    v8f acc = {0.f, 0.f, 0.f, 0.f, 0.f, 0.f, 0.f, 0.f};
#pragma unroll
    for (int kb = 0; kb < 4; ++kb) {
        int k = kb * 4 + 2 * half;  // this lane's two K rows of the B operand
        v2f b;
        b.x = t[base + k * sK + n16 * sN];
        b.y = t[base + (k + 1) * sK + n16 * sN];
        acc = __builtin_amdgcn_wmma_f32_16x16x4_f32(
            /*neg_a=*/false, aH[kb], /*neg_b=*/false, b,
            /*c_mod=*/(short)0, acc, /*reuse_a=*/false, /*reuse_b=*/false);
    }
#pragma unroll
    for (int r = 0; r < 8; ++r) {
        int m = 8 * half + r;  // C/D layout: VGPR r holds M = r (+8 for hi lanes)
        t[base + m * sK + n16 * sN] = acc[r];
    }
}

__global__ __launch_bounds__(TPB) void fwht_rot_kernel(
    const float* __restrict__ x, const float* __restrict__ signs,
    const int* __restrict__ perm, float* __restrict__ out) {
    __shared__ float raw[DIM];
    __shared__ float tt[TSIZE];

    const int tid = threadIdx.x;
    const size_t base = (size_t)blockIdx.x * DIM;

    // Phase A: fully coalesced row load into LDS
    const float4* xv = (const float4*)(x + base);
#pragma unroll
    for (int q = tid; q < DIM / 4; q += TPB) {
        ((float4*)raw)[q] = xv[q];
    }
    __syncthreads();

    // Phase B: permutation gather + sign flip, scatter into padded tensor
    //   linear j = a*256 + b*16 + c  ->  tt[(j>>4)*17 + (j&15)]
#pragma unroll
    for (int j = tid; j < DIM; j += TPB) {
        float v = raw[perm[j]] * signs[j];
        tt[(j >> 4) * TSTRIDE + (j & 15)] = v;
    }
    __syncthreads();

    // Per-lane constants + Hadamard A-operand (16x4 f32 layout:
    // lanes 0-15: M=lane, K = k0,k0+1; lanes 16-31: M=lane-16, K = k0+2,k0+3)
    const int lane = tid & 31;
    const int wave = tid >> 5;
    const int half = lane >> 4;
    const int n16 = lane & 15;
    v2f aH[4];
#pragma unroll
    for (int kb = 0; kb < 4; ++kb) {
        int k = kb * 4 + 2 * half;
        aH[kb].x = hval(n16, k);
        aH[kb].y = hval(n16, k + 1);
    }

    // Stage 1: transform along c.  S[k][n] = t[a][n][k]  (sK=1,  sN=17)
#pragma unroll
    for (int i = 0; i < 2; ++i) {
        int a = wave + 8 * i;
        had_tile(tt, a * (16 * TSTRIDE), 1, TSTRIDE, half, n16, aH);
    }
    __syncthreads();

    // Stage 2: transform along b.  S[k][n] = t[a][k][n]  (sK=17, sN=1)
#pragma unroll
    for (int i = 0; i < 2; ++i) {
        int a = wave + 8 * i;
        had_tile(tt, a * (16 * TSTRIDE), TSTRIDE, 1, half, n16, aH);
    }
    __syncthreads();

    // Stage 3: transform along a.  S[k][n] = t[k][b][n]  (sK=272, sN=1)
#pragma unroll
    for (int i = 0; i < 2; ++i) {
        int b = wave + 8 * i;
        had_tile(tt, b * TSTRIDE, 16 * TSTRIDE, 1, half, n16, aH);
    }
    __syncthreads();

    // Phase C: coalesced store with fused 1/sqrt(4096) scale
    const float scale = 0.015625f;  // 1/64
    float4* ov = (float4*)(out + base);
#pragma unroll
    for (int q = tid; q < DIM / 4; q += TPB) {
        int j = q * 4;
        int l = (j >> 4) * TSTRIDE + (j & 15);
        float4 v;
        v.x = tt[l + 0] * scale;
        v.y = tt[l + 1] * scale;
        v.z = tt[l + 2] * scale;
        v.w = tt[l + 3] * scale;
        ov[q] = v;
    }
}

extern "C" void kernel_launch(void* const* d_in, const int* in_sizes, int n_in,
                              void* d_out, int out_size, void* d_ws, size_t ws_size,
                              hipStream_t stream) {
    const float* x = (const float*)d_in[0];
    const float* signs = (const float*)d_in[1];
    const int* perm = (const int*)d_in[2];
    float* out = (float*)d_out;
    int rows = out_size / DIM;  // 4 * 4096 = 16384 rows
    fwht_rot_kernel<<<rows, TPB, 0, stream>>>(x, signs, perm, out);
}